// SpectralConvLayer_21131239096956
// MI455X (gfx1250) — compile-verified
//
#include <hip/hip_runtime.h>
#include <math.h>

// Problem constants (match reference)
#define NND  32768   // nodes
#define NB   64      // graphs
#define INC  256
#define OUTC 256
#define NSC  4
#define NH   8
#define HD   32      // head dim
#define QSCALE 0.17677669529663689f  // 32^-0.5

typedef float v2f __attribute__((ext_vector_type(2)));
typedef float v8f __attribute__((ext_vector_type(8)));

// fp32 WMMA: D(16x16,f32) = A(16x4,f32) * B(4x16,f32) + C
static __device__ __forceinline__ v8f wmma4(v2f a, v2f b, v8f c) {
  return __builtin_amdgcn_wmma_f32_16x16x4_f32(false, a, false, b, (short)0, c,
                                               false, false);
}

// wave-local LDS fence: cross-lane LDS visibility within one wave32
#define WAVE_FENCE() asm volatile("s_wait_dscnt 0" ::: "memory")

// ---------------------------------------------------------------- init / count
__global__ void k_init(int* counts, int* maxn) {
  int t = threadIdx.x;
  if (t < NB) counts[t] = 0;
  if (t == 0) *maxn = 0;
}

__global__ void k_count(const int* __restrict__ batch, int* counts) {
  int i = blockIdx.x * 256 + threadIdx.x;
  if (i < NND) atomicAdd(&counts[batch[i]], 1);
}

__global__ void k_scan(const int* __restrict__ counts, int* starts, int* maxn) {
  if (threadIdx.x == 0) {
    int acc = 0, mx = 0;
    for (int g = 0; g < NB; ++g) {
      starts[g] = acc;
      acc += counts[g];
      mx = max(mx, counts[g]);
    }
    *maxn = mx;
  }
}

// ------------------------------------------------- stage 1: per-scale linears
// h[n, s*64+o] = sum_i (x[n,i]+spectral[s,n,i]) * Ws[s,i,o] + bs[s,o]
__global__ void k_scale_linear(const float* __restrict__ x,
                               const float* __restrict__ spec,
                               const float* __restrict__ Ws,
                               const float* __restrict__ bs,
                               float* __restrict__ h) {
  const int lane = threadIdx.x & 31, wave = threadIdx.x >> 5;
  const int tile = blockIdx.x * 8 + wave;
  const int TPC = OUTC / 16;              // 16 column tiles
  const int tr = tile / TPC, tc = tile % TPC;
  if (tr >= NND / 16) return;
  const int s = tc >> 2;                  // scale
  const int oc0 = (tc & 3) * 16;          // column in this scale's 64-wide out
  const int hi = lane >> 4, lx = lane & 15;
  const int arow = tr * 16 + lx;
  const float* xb = x + (size_t)arow * INC;
  const float* sb = spec + (size_t)s * NND * INC + (size_t)arow * INC;
  const float* wb = Ws + (size_t)s * INC * 64;
  v8f c = {};
  for (int k = 0; k < INC; k += 4) {
    int ka = k + 2 * hi;                  // f32 A-layout: hi half holds K+2,K+3
    v2f a; a.x = xb[ka] + sb[ka];  a.y = xb[ka + 1] + sb[ka + 1];
    v2f b; b.x = wb[(size_t)ka * 64 + oc0 + lx];
           b.y = wb[(size_t)(ka + 1) * 64 + oc0 + lx];
    c = wmma4(a, b, c);
  }
  const float bias = bs[s * 64 + oc0 + lx];
  float* hb = h + (size_t)(tr * 16) * OUTC + s * 64 + oc0 + lx;
  for (int r = 0; r < 8; ++r)
    hb[(size_t)(r + 8 * hi) * OUTC] = c[r] + bias;
}

// ---------------------------------------------------------- stage 2: QKV GEMM
// qkv[n,e] = sum_j h[n,j]*W_in[e*256+j] + b_in[e]; q pre-scaled by dh^-0.5
__global__ void k_qkv(const float* __restrict__ h,
                      const float* __restrict__ Win,
                      const float* __restrict__ bin,
                      float* __restrict__ qb, float* __restrict__ kb,
                      float* __restrict__ vb) {
  const int lane = threadIdx.x & 31, wave = threadIdx.x >> 5;
  const int tile = blockIdx.x * 8 + wave;
  const int TPC = 3 * OUTC / 16;          // 48 column tiles
  const int tr = tile / TPC, tc = tile % TPC;
  if (tr >= NND / 16) return;
  const int hi = lane >> 4, lx = lane & 15;
  const int e = tc * 16 + lx;
  const float* ab = h + (size_t)(tr * 16 + lx) * INC;
  const float* wb = Win + (size_t)e * INC;
  v8f c = {};
  for (int k = 0; k < INC; k += 4) {
    int ka = k + 2 * hi;
    v2f a; a.x = ab[ka]; a.y = ab[ka + 1];
    v2f b; b.x = wb[ka]; b.y = wb[ka + 1];
    c = wmma4(a, b, c);
  }
  const float bias = bin[e];
  float scale; float* dst; int ec;
  if (e < OUTC)          { dst = qb; ec = e;           scale = QSCALE; }
  else if (e < 2 * OUTC) { dst = kb; ec = e - OUTC;    scale = 1.f; }
  else                   { dst = vb; ec = e - 2*OUTC;  scale = 1.f; }
  for (int r = 0; r < 8; ++r)
    dst[(size_t)(tr * 16 + r + 8 * hi) * OUTC + ec] = (c[r] + bias) * scale;
}

// --------------------------------------- stage 3: ragged attention (two-pass)
__global__ void k_attn(const float* __restrict__ qb,
                       const float* __restrict__ kbuf,
                       const float* __restrict__ vbuf,
                       const int* __restrict__ counts,
                       const int* __restrict__ starts,
                       const int* __restrict__ maxnp,
                       float* __restrict__ ctx) {
  __shared__ float Qs[16 * 33];
  __shared__ float Ss[16 * 17];
  __shared__ float rowm[16];
  __shared__ float rden[16];
  const int lane = threadIdx.x;
  const int hi = lane >> 4, lx = lane & 15;
  const int wv = blockIdx.x & 7;
  const int hh = (blockIdx.x >> 3) & (NH - 1);
  const int b  = blockIdx.x >> 6;
  const int cnt = counts[b];
  if (cnt <= 0) return;
  const int gs = starts[b];
  const int npad = *maxnp - cnt;          // unmasked zero-pad keys
  const int chan = hh * HD;
  const int nkt = (cnt + 15) >> 4;

  for (int qt = wv; qt * 16 < cnt; qt += 8) {
    // stage Q tile [16 x 32] in LDS (coalesced)
    for (int idx = lane; idx < 16 * HD; idx += 32) {
      int r = idx >> 5, d = idx & 31, n = qt * 16 + r;
      Qs[r * 33 + d] = (n < cnt) ? qb[(size_t)(gs + n) * OUTC + chan + d] : 0.f;
    }
    WAVE_FENCE();

    // ---- pass 1: row max over real keys (pads contribute score 0)
    float m = -3.0e38f;
    for (int kt = 0; kt < nkt; ++kt) {
      const int kb0 = kt * 16;
      const int jmax = min(16, cnt - kb0);
      const int j = kb0 + lx;
      const bool jv = (j < cnt);
      const float* kp = kbuf + (size_t)(gs + j) * OUTC + chan;
      v8f c = {};
      for (int st = 0; st < 8; ++st) {
        int kk = st * 4 + 2 * hi;
        v2f a; a.x = Qs[lx * 33 + kk]; a.y = Qs[lx * 33 + kk + 1];
        v2f bv; bv.x = jv ? kp[kk] : 0.f; bv.y = jv ? kp[kk + 1] : 0.f;
        c = wmma4(a, bv, c);
      }
      for (int r = 0; r < 8; ++r) Ss[(r + 8 * hi) * 17 + lx] = c[r];
      WAVE_FENCE();
      if (lane < 16)
        for (int jj = 0; jj < jmax; ++jj) m = fmaxf(m, Ss[lane * 17 + jj]);
      WAVE_FENCE();
    }
    if (npad > 0) m = fmaxf(m, 0.f);
    if (lane < 16) rowm[lane] = m;
    WAVE_FENCE();

    // ---- pass 2: P = exp(S - rm), den accumulation, ctx += P @ V
    v8f c0 = {}, c1 = {};
    float den = 0.f;
    for (int kt = 0; kt < nkt; ++kt) {
      const int kb0 = kt * 16;
      const int jmax = min(16, cnt - kb0);
      const int j = kb0 + lx;
      const bool jv = (j < cnt);
      const float* kp = kbuf + (size_t)(gs + j) * OUTC + chan;
      v8f c = {};
      for (int st = 0; st < 8; ++st) {
        int kk = st * 4 + 2 * hi;
        v2f a; a.x = Qs[lx * 33 + kk]; a.y = Qs[lx * 33 + kk + 1];
        v2f bv; bv.x = jv ? kp[kk] : 0.f; bv.y = jv ? kp[kk + 1] : 0.f;
        c = wmma4(a, bv, c);
      }
      for (int r = 0; r < 8; ++r) Ss[(r + 8 * hi) * 17 + lx] = c[r];
      WAVE_FENCE();
      if (lane < 16) {
        const float rm = rowm[lane];
        for (int jj = 0; jj < 16; ++jj) {
          float p = (jj < jmax) ? __expf(Ss[lane * 17 + jj] - rm) : 0.f;
          Ss[lane * 17 + jj] = p;
          den += p;
        }
      }
      WAVE_FENCE();
      for (int st = 0; st < 4; ++st) {
        int kk = st * 4 + 2 * hi;
        v2f a; a.x = Ss[lx * 17 + kk]; a.y = Ss[lx * 17 + kk + 1];
        const int n0 = kb0 + kk;
        const float* vp0 = vbuf + (size_t)(gs + n0) * OUTC + chan;
        const float* vp1 = vp0 + OUTC;
        const bool v0 = (n0 < cnt), v1 = (n0 + 1 < cnt);
        v2f b0, b1;
        b0.x = v0 ? vp0[lx] : 0.f;      b0.y = v1 ? vp1[lx] : 0.f;
        b1.x = v0 ? vp0[16 + lx] : 0.f; b1.y = v1 ? vp1[16 + lx] : 0.f;
        c0 = wmma4(a, b0, c0);
        c1 = wmma4(a, b1, c1);
      }
      WAVE_FENCE();
    }
    if (lane < 16) {
      float d = den;
      if (npad > 0) d += (float)npad * __expf(-rowm[lane]);  // pad keys
      rden[lane] = d;
    }
    WAVE_FENCE();
    for (int r = 0; r < 8; ++r) {
      const int row = r + 8 * hi;
      const int n = qt * 16 + row;
      if (n < cnt) {
        const float invd = 1.f / rden[row];
        const size_t o = (size_t)(gs + n) * OUTC + chan;
        ctx[o + lx]      = c0[r] * invd;
        ctx[o + 16 + lx] = c1[r] * invd;
      }
    }
    WAVE_FENCE();
  }
}

// ------------------------------------------------ stage 4: output projection
__global__ void k_outproj(const float* __restrict__ ctxm,
                          const float* __restrict__ Wout,
                          const float* __restrict__ bout,
                          float* __restrict__ out) {
  const int lane = threadIdx.x & 31, wave = threadIdx.x >> 5;
  const int tile = blockIdx.x * 8 + wave;
  const int TPC = OUTC / 16;
  const int tr = tile / TPC, tc = tile % TPC;
  if (tr >= NND / 16) return;
  const int hi = lane >> 4, lx = lane & 15;
  const int e = tc * 16 + lx;
  const float* ab = ctxm + (size_t)(tr * 16 + lx) * OUTC;
  const float* wb = Wout + (size_t)e * OUTC;
  v8f c = {};
  for (int k = 0; k < OUTC; k += 4) {
    int ka = k + 2 * hi;
    v2f a; a.x = ab[ka]; a.y = ab[ka + 1];
    v2f b; b.x = wb[ka]; b.y = wb[ka + 1];
    c = wmma4(a, b, c);
  }
  const float bias = bout[e];
  for (int r = 0; r < 8; ++r)
    out[(size_t)(tr * 16 + r + 8 * hi) * OUTC + e] = c[r] + bias;
}

// ------------------------------------- stage 5: BatchNorm (deterministic) + ReLU
__global__ void k_bnpart(const float* __restrict__ h2, float* psum, float* psq) {
  const int t = threadIdx.x, bidx = blockIdx.x;
  const int RPB = NND / 256;
  float s = 0.f, s2 = 0.f;
  for (int r = 0; r < RPB; ++r) {
    float v = h2[(size_t)(bidx * RPB + r) * OUTC + t];
    s += v; s2 += v * v;
  }
  psum[bidx * OUTC + t] = s;
  psq[bidx * OUTC + t]  = s2;
}

__global__ void k_bnreduce(const float* __restrict__ psum,
                           const float* __restrict__ psq,
                           float* bnsum, float* bnsq) {
  const int t = threadIdx.x;
  float s = 0.f, s2 = 0.f;
  for (int b = 0; b < 256; ++b) { s += psum[b * OUTC + t]; s2 += psq[b * OUTC + t]; }
  bnsum[t] = s; bnsq[t] = s2;
}

__global__ void k_bnapply(const float* __restrict__ bnsum,
                          const float* __restrict__ bnsq,
                          const float* __restrict__ gamma,
                          const float* __restrict__ beta,
                          float* out) {
  const size_t idx = (size_t)blockIdx.x * 256 + threadIdx.x;
  const int ch = (int)(idx & (OUTC - 1));
  const float invn = 1.f / (float)NND;
  const float mean = bnsum[ch] * invn;
  const float var  = bnsq[ch] * invn - mean * mean;
  const float v = out[idx];
  const float y = (v - mean) * rsqrtf(var + 1e-5f) * gamma[ch] + beta[ch];
  out[idx] = fmaxf(y, 0.f);
}

// -------------------------------------------------------------------- launch
extern "C" void kernel_launch(void* const* d_in, const int* in_sizes, int n_in,
                              void* d_out, int out_size, void* d_ws, size_t ws_size,
                              hipStream_t stream) {
  (void)in_sizes; (void)n_in; (void)out_size; (void)ws_size;
  const float* x        = (const float*)d_in[0];
  const float* spectral = (const float*)d_in[1];
  const float* Ws       = (const float*)d_in[2];
  const float* bs       = (const float*)d_in[3];
  const float* W_in     = (const float*)d_in[4];
  const float* b_in     = (const float*)d_in[5];
  const float* W_out    = (const float*)d_in[6];
  const float* b_out    = (const float*)d_in[7];
  const float* gamma    = (const float*)d_in[8];
  const float* beta     = (const float*)d_in[9];
  const int*   batch    = (const int*)d_in[10];
  float* out = (float*)d_out;

  char* ws = (char*)d_ws;
  int*   counts = (int*)(ws + 0);
  int*   starts = (int*)(ws + 256);
  int*   maxn   = (int*)(ws + 512);
  float* bnsum  = (float*)(ws + 1024);
  float* bnsq   = (float*)(ws + 2048);
  float* psum   = (float*)(ws + 4096);
  float* psq    = (float*)(ws + 4096 + 256 * OUTC * sizeof(float));
  const size_t MAT = (size_t)NND * OUTC * sizeof(float);
  char* big = ws + (1u << 20);
  float* h   = (float*)(big + 0 * MAT);   // h, later reused as ctx
  float* qb  = (float*)(big + 1 * MAT);
  float* kb  = (float*)(big + 2 * MAT);
  float* vb  = (float*)(big + 3 * MAT);
  float* ctx = h;                         // safe: h dead after k_qkv

  k_init<<<1, 256, 0, stream>>>(counts, maxn);
  k_count<<<NND / 256, 256, 0, stream>>>(batch, counts);
  k_scan<<<1, 32, 0, stream>>>(counts, starts, maxn);

  // GEMM grids: one 16x16 tile per wave, 8 waves per block
  k_scale_linear<<<(NND / 16) * (OUTC / 16) / 8, 256, 0, stream>>>(x, spectral, Ws, bs, h);
  k_qkv<<<(NND / 16) * (3 * OUTC / 16) / 8, 256, 0, stream>>>(h, W_in, b_in, qb, kb, vb);
  k_attn<<<NB * NH * 8, 32, 0, stream>>>(qb, kb, vb, counts, starts, maxn, ctx);
  k_outproj<<<(NND / 16) * (OUTC / 16) / 8, 256, 0, stream>>>(ctx, W_out, b_out, out);

  k_bnpart<<<256, 256, 0, stream>>>(out, psum, psq);
  k_bnreduce<<<1, 256, 0, stream>>>(psum, psq, bnsum, bnsq);
  k_bnapply<<<(NND * OUTC) / 256, 256, 0, stream>>>(bnsum, bnsq, gamma, beta, out);
}